// GATModel_55860344652097
// MI455X (gfx1250) — compile-verified
//
#include <hip/hip_runtime.h>
#include <math.h>

typedef __attribute__((ext_vector_type(2))) float v2f;
typedef __attribute__((ext_vector_type(8))) float v8f;

#define NEG_SLOPE 0.2f
#define GAT_EPS 1e-16f
#define ORD_NEG_INF ((int)0x807FFFFF)   // ordered-int encoding of -inf

// ---- ordered-int <-> float monotone bijection (for float atomicMax) ----
__device__ __forceinline__ int ordf(float f) {
  int b = __float_as_int(f);
  return b >= 0 ? b : (b ^ 0x7fffffff);
}
__device__ __forceinline__ float unordf(int b) {
  return __int_as_float(b >= 0 ? b : (b ^ 0x7fffffff));
}
__device__ __forceinline__ float lrelu(float x) {
  return x > 0.f ? x : NEG_SLOPE * x;
}

// =====================================================================
// H[N,128] = X[N,128] @ W[128,128]   (fp32 WMMA 16x16x4, wave32)
// Block = 256 threads = 8 waves. Block b owns rows [16b,16b+16);
// wave w owns cols [16w,16w+16). K-loop: 32 wmma steps of K=4.
// A frag (16x4 f32): lane<16 rows M=lane, K=kk+{0,1}; lane>=16 K=kk+{2,3}.
// B frag (4x16 f32): vgpr i holds row K=kk+2*half+i, col = lane&15.
// C/D (16x16 f32): vgpr i -> row i+8*half, col = lane&15.
// =====================================================================
__global__ void gemm128_wmma(const float* __restrict__ X,
                             const float* __restrict__ W,
                             float* __restrict__ H) {
  const int wave = threadIdx.x >> 5;
  const int lane = threadIdx.x & 31;
  const int half = lane >> 4;
  const int l16  = lane & 15;
  const int row0 = blockIdx.x << 4;
  const int col0 = wave << 4;

  const float* xrow = X + (size_t)(row0 + l16) * 128;
  const float* wcol = W + col0 + l16;

  v8f acc = {0.f, 0.f, 0.f, 0.f, 0.f, 0.f, 0.f, 0.f};
#pragma unroll
  for (int kk = 0; kk < 128; kk += 4) {
    const int k = kk + 2 * half;
    v2f a, b;
    a.x = xrow[k];
    a.y = xrow[k + 1];
    b.x = wcol[(size_t)k * 128];
    b.y = wcol[(size_t)(k + 1) * 128];
    acc = __builtin_amdgcn_wmma_f32_16x16x4_f32(false, a, false, b,
                                                (short)0, acc, false, false);
  }
  float* hrow = H + (size_t)(row0 + 8 * half) * 128 + col0 + l16;
#pragma unroll
  for (int i = 0; i < 8; ++i) hrow[(size_t)i * 128] = acc[i];
}

// ---- attention logits: al_s[n,h] = <H[n,h,:], a_src[h,:]> ----
__global__ void attn_logits(const float* __restrict__ Hf,
                            const float* __restrict__ asv,
                            const float* __restrict__ adv,
                            float* __restrict__ als, float* __restrict__ ald,
                            int N) {
  int t = blockIdx.x * blockDim.x + threadIdx.x;
  if (t >= N * 2) return;
  int n = t >> 1, h = t & 1;
  const float* hp = Hf + (size_t)n * 128 + h * 64;
  const float* ap = asv + h * 64;
  const float* dp = adv + h * 64;
  float s = 0.f, d = 0.f;
#pragma unroll 8
  for (int c = 0; c < 64; ++c) {
    float v = hp[c];
    s += v * ap[c];
    d += v * dp[c];
  }
  als[t] = s;
  ald[t] = d;
}

__global__ void fill_f32(float* __restrict__ p, float v, int n) {
  int t = blockIdx.x * blockDim.x + threadIdx.x;
  if (t < n) p[t] = v;
}
__global__ void fill_i32(int* __restrict__ p, int v, int n) {
  int t = blockIdx.x * blockDim.x + threadIdx.x;
  if (t < n) p[t] = v;
}

__device__ __forceinline__ void edge_sd(const int* __restrict__ ei, int E,
                                        int e, int& s, int& d) {
  if (e < E) { s = ei[e]; d = ei[E + e]; }
  else       { s = e - E; d = e - E; }          // self loops appended
}

// ---- pass 1: segment max of leaky_relu(al_s[src]+al_d[dst]) over dst ----
__global__ void edge_max(const int* __restrict__ ei, int E, int Etot,
                         const float* __restrict__ als,
                         const float* __restrict__ ald,
                         int* __restrict__ mord) {
  int t = blockIdx.x * blockDim.x + threadIdx.x;
  if (t >= Etot * 2) return;
  int e = t >> 1, h = t & 1, s, d;
  edge_sd(ei, E, e, s, d);
  float x = lrelu(als[s * 2 + h] + ald[d * 2 + h]);
  atomicMax(&mord[d * 2 + h], ordf(x));
}

// ---- pass 2: denom[dst,h] = sum exp(e - max) ----
__global__ void edge_sumexp(const int* __restrict__ ei, int E, int Etot,
                            const float* __restrict__ als,
                            const float* __restrict__ ald,
                            const int* __restrict__ mord,
                            float* __restrict__ denom) {
  int t = blockIdx.x * blockDim.x + threadIdx.x;
  if (t >= Etot * 2) return;
  int e = t >> 1, h = t & 1, s, d;
  edge_sd(ei, E, e, s, d);
  float x = lrelu(als[s * 2 + h] + ald[d * 2 + h]);
  float w = __expf(x - unordf(mord[d * 2 + h]));
  atomicAdd(&denom[d * 2 + h], w);
}

// ---- pass 3: out[dst] += alpha * H[src]  (one wave per edge, float4/lane) ----
__global__ void edge_aggregate(const int* __restrict__ ei, int E, int Etot,
                               const float* __restrict__ als,
                               const float* __restrict__ ald,
                               const int* __restrict__ mord,
                               const float* __restrict__ denom,
                               const float* __restrict__ Hf,
                               float* __restrict__ out) {
  int t = blockIdx.x * blockDim.x + threadIdx.x;
  int e = t >> 5;
  if (e >= Etot) return;
  int lane = t & 31;
  int h = lane >> 4;          // channels 4*lane..4*lane+3 all in head lane/16
  int s, d;
  edge_sd(ei, E, e, s, d);
  float x = lrelu(als[s * 2 + h] + ald[d * 2 + h]);
  float alpha = __expf(x - unordf(mord[d * 2 + h])) / (denom[d * 2 + h] + GAT_EPS);
  const float4 hv = *(const float4*)(Hf + (size_t)s * 128 + lane * 4);
  float* op = out + (size_t)d * 128 + lane * 4;
  atomicAdd(op + 0, alpha * hv.x);
  atomicAdd(op + 1, alpha * hv.y);
  atomicAdd(op + 2, alpha * hv.z);
  atomicAdd(op + 3, alpha * hv.w);
}

// ---- out = relu(out + bias), float4 per thread ----
__global__ void bias_relu(float* __restrict__ p, const float* __restrict__ b,
                          int N) {
  int t = blockIdx.x * blockDim.x + threadIdx.x;
  if (t >= N * 32) return;
  int c4 = (t & 31) * 4;
  float4 v = *(float4*)(p + (size_t)t * 4);
  v.x = fmaxf(v.x + b[c4 + 0], 0.f);
  v.y = fmaxf(v.y + b[c4 + 1], 0.f);
  v.z = fmaxf(v.z + b[c4 + 2], 0.f);
  v.w = fmaxf(v.w + b[c4 + 3], 0.f);
  *(float4*)(p + (size_t)t * 4) = v;
}

// ---- global mean pool: sums + counts ----
__global__ void pool_sum(const float* __restrict__ Hf,
                         const int* __restrict__ batch,
                         float* __restrict__ pooled, int N) {
  int t = blockIdx.x * blockDim.x + threadIdx.x;
  if (t >= N * 32) return;
  int n = t >> 5, c4 = (t & 31) * 4;
  int g = batch[n];
  const float4 v = *(const float4*)(Hf + (size_t)n * 128 + c4);
  float* pp = pooled + (size_t)g * 128 + c4;
  atomicAdd(pp + 0, v.x);
  atomicAdd(pp + 1, v.y);
  atomicAdd(pp + 2, v.z);
  atomicAdd(pp + 3, v.w);
}
__global__ void pool_cnt(const int* __restrict__ batch, float* __restrict__ cnt,
                         int N) {
  int n = blockIdx.x * blockDim.x + threadIdx.x;
  if (n < N) atomicAdd(&cnt[batch[n]], 1.f);
}

// ---- head: out[g] = (pooled[g]/max(cnt,1)) . Wh + bh ----
__global__ void head_kernel(const float* __restrict__ pooled,
                            const float* __restrict__ cnt,
                            const float* __restrict__ Wh,
                            const float* __restrict__ bh,
                            float* __restrict__ out) {
  __shared__ float red[128];
  int g = blockIdx.x, c = threadIdx.x;
  float cv = fmaxf(cnt[g], 1.f);
  red[c] = (pooled[(size_t)g * 128 + c] / cv) * Wh[c];
  __syncthreads();
  for (int s = 64; s > 0; s >>= 1) {
    if (c < s) red[c] += red[c + s];
    __syncthreads();
  }
  if (c == 0) out[g] = red[0] + bh[0];
}

static inline int cdiv(long long a, long long b) { return (int)((a + b - 1) / b); }

extern "C" void kernel_launch(void* const* d_in, const int* in_sizes, int n_in,
                              void* d_out, int out_size, void* d_ws, size_t ws_size,
                              hipStream_t stream) {
  const float* x    = (const float*)d_in[0];
  const int*   ei   = (const int*)d_in[1];
  const int*   batch= (const int*)d_in[2];
  const float* W1   = (const float*)d_in[3];
  const float* as1  = (const float*)d_in[4];
  const float* ad1  = (const float*)d_in[5];
  const float* b1   = (const float*)d_in[6];
  const float* W2   = (const float*)d_in[7];
  const float* as2  = (const float*)d_in[8];
  const float* ad2  = (const float*)d_in[9];
  const float* b2   = (const float*)d_in[10];
  const float* Wh   = (const float*)d_in[11];
  const float* bh   = (const float*)d_in[12];
  float* out = (float*)d_out;

  const int N    = in_sizes[0] / 128;
  const int E    = in_sizes[1] / 2;
  const int Etot = E + N;           // reference appends N self-loops
  const int G    = out_size;

  // ---- workspace carve-up ----
  char* ws = (char*)d_ws;
  float* bufA   = (float*)ws; ws += (size_t)N * 128 * sizeof(float); // GEMM out / gather src
  float* bufB   = (float*)ws; ws += (size_t)N * 128 * sizeof(float); // aggregation target
  float* als    = (float*)ws; ws += (size_t)N * 2 * sizeof(float);
  float* ald    = (float*)ws; ws += (size_t)N * 2 * sizeof(float);
  int*   mord   = (int*)ws;   ws += (size_t)N * 2 * sizeof(int);
  float* den    = (float*)ws; ws += (size_t)N * 2 * sizeof(float);
  float* pooled = (float*)ws; ws += (size_t)G * 128 * sizeof(float);
  float* cnt    = (float*)ws; ws += (size_t)G * sizeof(float);
  (void)ws_size; (void)n_in;

  const int BT = 256;
  const float* layer_in = x;

  for (int layer = 0; layer < 2; ++layer) {
    const float* W  = layer ? W2 : W1;
    const float* av = layer ? as2 : as1;
    const float* dv = layer ? ad2 : ad1;
    const float* bv = layer ? b2 : b1;

    // h = in @ W  (N divisible by 16 for these shapes; EXEC all-ones at WMMA)
    gemm128_wmma<<<N / 16, 256, 0, stream>>>(layer_in, W, bufA);

    attn_logits<<<cdiv((long long)N * 2, BT), BT, 0, stream>>>(bufA, av, dv,
                                                               als, ald, N);
    fill_i32<<<cdiv((long long)N * 2, BT), BT, 0, stream>>>(mord, ORD_NEG_INF, N * 2);
    fill_f32<<<cdiv((long long)N * 2, BT), BT, 0, stream>>>(den, 0.f, N * 2);
    fill_f32<<<cdiv((long long)N * 128, BT), BT, 0, stream>>>(bufB, 0.f, N * 128);

    edge_max<<<cdiv((long long)Etot * 2, BT), BT, 0, stream>>>(ei, E, Etot,
                                                               als, ald, mord);
    edge_sumexp<<<cdiv((long long)Etot * 2, BT), BT, 0, stream>>>(ei, E, Etot,
                                                                  als, ald, mord, den);
    edge_aggregate<<<cdiv((long long)Etot * 32, BT), BT, 0, stream>>>(
        ei, E, Etot, als, ald, mord, den, bufA, bufB);

    bias_relu<<<cdiv((long long)N * 32, BT), BT, 0, stream>>>(bufB, bv, N);
    layer_in = bufB;   // next GEMM reads bufB, writes bufA (safe ping-pong)
  }

  // ---- global mean pool + linear head ----
  fill_f32<<<cdiv((long long)G * 128, BT), BT, 0, stream>>>(pooled, 0.f, G * 128);
  fill_f32<<<cdiv((long long)G, BT), BT, 0, stream>>>(cnt, 0.f, G);
  pool_sum<<<cdiv((long long)N * 32, BT), BT, 0, stream>>>(bufB, batch, pooled, N);
  pool_cnt<<<cdiv((long long)N, BT), BT, 0, stream>>>(batch, cnt, N);
  head_kernel<<<G, 128, 0, stream>>>(pooled, cnt, Wh, bh, out);
}